// DiffeqSolver_84859963834845
// MI455X (gfx1250) — compile-verified
//
#include <hip/hip_runtime.h>

// Neural-ODE RK4 on MI455X (gfx1250).
// N=4096 rows, D=64, H=256, T=200. Rows are independent -> one wave32 owns
// 16 rows for the whole time loop; state stays in VGPRs (A-fragment layout),
// weights stay in LDS pre-swizzled into B-fragment order (ds_load_2addr_b64
// serves two WMMA operands per instruction, bank-conflict-free), only the
// trajectory streams to HBM (210 MB). All matrix math is
// V_WMMA_F32_16X16X4_F32 (full fp32 to match the f32 reference numerics).
// Accumulators zero-init (inline-0 SRC2 WMMA form); biases folded into the
// tanh fma / writeback adds.

#define N_PTS   4096
#define D_DIM   64
#define H_DIM   256
#define T_STEPS 200

#define TANH_K  2.8853900817779268f   // 2*log2(e)

typedef float v2f __attribute__((ext_vector_type(2)));
typedef float v8f __attribute__((ext_vector_type(8)));

// Assumed wave32 f32 fragment layouts (per ISA 7.12.2):
//  A (16x4):  VGPR j, lane L: m = L&15, k = 2*(L>>4) + j
//  B (4x16):  VGPR j, lane L: n = L&15, k = 2*(L>>4) + j
//  C/D(16x16):VGPR v, lane L: m = v + 8*(L>>4), n = L&15

__device__ __forceinline__ v8f wmma_f32(v2f a, v2f b, v8f c) {
    return __builtin_amdgcn_wmma_f32_16x16x4_f32(
        /*neg_a=*/false, a, /*neg_b=*/false, b,
        /*c_mod=*/(short)0, c, /*reuse_a=*/false, /*reuse_b=*/false);
}

// tanh(x + b) with pre-scaled bias bK = b*TANH_K:
// tanh(u) = 1 - 2/(exp(2u)+1); fma + v_exp_f32 + add + v_rcp_f32 + fma.
__device__ __forceinline__ float fast_tanh_biased(float x, float bK) {
    float e = __builtin_amdgcn_exp2f(__builtin_fmaf(x, TANH_K, bK));
    float r = __builtin_amdgcn_rcpf(e + 1.0f);
    return __builtin_fmaf(r, -2.0f, 1.0f);
}

// f(y) = tanh(y*W1 + b1)*W2 + b2 for one wave's 16 rows.
// yA: 16 x v2f, A-fragment layout of y[16,64]. kA: result, same layout.
// W1f/W2f: weights pre-swizzled into per-lane B-fragment pairs.
// scr: 1024-float wave-private LDS scratch (layout conversion).
__device__ __forceinline__ void eval_f(const v2f* __restrict__ yA,
                                       v2f* __restrict__ kA,
                                       const float* __restrict__ W1f,
                                       const float* __restrict__ W2f,
                                       const float* __restrict__ b1s,
                                       const float* __restrict__ b2s,
                                       float* __restrict__ scr,
                                       int lane, int lo, int hi) {
    // C2 accumulators: output [16,64] = 4 tiles of 16x16, zero-init
    // (first WMMA of each chain uses the inline-0 C operand); b2 added at
    // writeback.
    v8f c2[4];
#pragma unroll
    for (int nt = 0; nt < 4; ++nt) c2[nt] = (v8f){};

    // Hidden tiles in pairs: two independent WMMA accumulator chains (ILP).
#pragma unroll 1
    for (int hc = 0; hc < 16; hc += 2) {
        v8f c1a = (v8f){};
        v8f c1b = (v8f){};
        float bK0 = b1s[hc * 16 + lo] * TANH_K;        // bias folded into tanh
        float bK1 = b1s[(hc + 1) * 16 + lo] * TANH_K;
        // GEMM1: C1 = y[16,64] * W1[:, hc*16 .. hc*16+31]
#pragma unroll
        for (int kc = 0; kc < 16; ++kc) {
            v2f a  = yA[kc];
            v2f w0 = *(const v2f*)(W1f + ((kc * 16 + hc) * 32 + lane) * 2);
            v2f w1 = *(const v2f*)(W1f + ((kc * 16 + hc + 1) * 32 + lane) * 2);
            c1a = wmma_f32(a, w0, c1a);
            c1b = wmma_f32(a, w1, c1b);
        }
        // tanh(+bias), then park both 16x16 tiles in LDS (C layout -> row major).
#pragma unroll
        for (int v = 0; v < 8; ++v) {
            scr[(v + 8 * hi) * 16 + lo]       = fast_tanh_biased(c1a[v], bK0);
            scr[256 + (v + 8 * hi) * 16 + lo] = fast_tanh_biased(c1b[v], bK1);
        }
        // GEMM2: re-read h tiles as A fragments, accumulate into C2.
#pragma unroll
        for (int half = 0; half < 2; ++half) {
            const float* ht = scr + half * 256;
            int hh = hc + half;
#pragma unroll
            for (int s = 0; s < 4; ++s) {
                v2f ah = *(const v2f*)(ht + lo * 16 + 4 * s + 2 * hi);
#pragma unroll
                for (int nt = 0; nt < 4; ++nt) {
                    v2f w = *(const v2f*)(W2f + (((hh * 4 + s) * 4 + nt) * 32 + lane) * 2);
                    c2[nt] = wmma_f32(ah, w, c2[nt]);
                }
            }
        }
    }

    // Convert C2 (C layout) -> kA (A layout) through LDS; add b2 here.
#pragma unroll
    for (int nt = 0; nt < 4; ++nt) {
        float bb = b2s[nt * 16 + lo];
#pragma unroll
        for (int v = 0; v < 8; ++v)
            scr[(v + 8 * hi) * 64 + nt * 16 + lo] = c2[nt][v] + bb;
    }
#pragma unroll
    for (int kc = 0; kc < 16; ++kc)
        kA[kc] = *(const v2f*)(scr + lo * 64 + 4 * kc + 2 * hi);
}

// Write y (A layout) to out[(R..R+15), t, :] via LDS for coalescing:
// each row is a contiguous 256B chunk; 8 b128 stores per wave.
__device__ __forceinline__ void store_traj(float* __restrict__ out,
                                           const v2f* __restrict__ yA,
                                           float* __restrict__ scr,
                                           int R, int t, int lane, int lo, int hi) {
#pragma unroll
    for (int kc = 0; kc < 16; ++kc)
        *(v2f*)(scr + lo * 64 + 4 * kc + 2 * hi) = yA[kc];
#pragma unroll
    for (int i = 0; i < 8; ++i) {
        int f4  = i * 32 + lane;
        int row = f4 >> 4;
        int c4  = (f4 & 15) * 4;
        float4 v = *(const float4*)(scr + row * 64 + c4);
        *(float4*)(out + ((size_t)(R + row) * T_STEPS + t) * D_DIM + c4) = v;
    }
}

__global__ void __launch_bounds__(64)
node_rk4_kernel(const float* __restrict__ y0, const float* __restrict__ ts,
                const float* __restrict__ W1, const float* __restrict__ b1,
                const float* __restrict__ W2, const float* __restrict__ b2,
                float* __restrict__ out) {
    extern __shared__ float smem[];
    float* W1f = smem;                     // 16384, B-fragment swizzled
    float* W2f = W1f + D_DIM * H_DIM;      // 16384, B-fragment swizzled
    float* b1s = W2f + H_DIM * D_DIM;      // 256
    float* b2s = b1s + H_DIM;              // 64
    float* scr = b2s + D_DIM;              // 2 waves * 1024

    // Cooperative weight load, swizzling into B-fragment order so each lane's
    // WMMA operand pair is 8 contiguous bytes (ds_load_b64 / 2addr_b64 later).
    // W1f[((kc*16+hc)*32+L)*2+j] = W1[(4kc+2(L>>4)+j) * H + hc*16+(L&15)]
    for (int idx = threadIdx.x; idx < D_DIM * H_DIM; idx += blockDim.x) {
        int j  = idx & 1;
        int L  = (idx >> 1) & 31;
        int hc = (idx >> 6) & 15;
        int kc = idx >> 10;
        int r  = 4 * kc + 2 * (L >> 4) + j;
        int c  = hc * 16 + (L & 15);
        W1f[idx] = W1[r * H_DIM + c];
    }
    // W2f[(((hc*4+s)*4+nt)*32+L)*2+j] = W2[(hc*16+4s+2(L>>4)+j) * D + nt*16+(L&15)]
    for (int idx = threadIdx.x; idx < H_DIM * D_DIM; idx += blockDim.x) {
        int j  = idx & 1;
        int L  = (idx >> 1) & 31;
        int nt = (idx >> 6) & 3;
        int s  = (idx >> 8) & 3;
        int hc = idx >> 10;
        int k  = hc * 16 + 4 * s + 2 * (L >> 4) + j;
        int d  = nt * 16 + (L & 15);
        W2f[idx] = W2[k * D_DIM + d];
    }
    for (int i = threadIdx.x; i < H_DIM; i += blockDim.x) b1s[i] = b1[i];
    for (int i = threadIdx.x; i < D_DIM; i += blockDim.x) b2s[i] = b2[i];
    __syncthreads();

    const int lane = threadIdx.x & 31;
    const int wave = threadIdx.x >> 5;
    const int lo = lane & 15;
    const int hi = lane >> 4;
    float* myScr = scr + wave * 1024;      // wave-private: no barriers needed
    const int R = blockIdx.x * 32 + wave * 16;

    // Load y0 rows into A-fragment registers.
    v2f yA[16];
#pragma unroll
    for (int kc = 0; kc < 16; ++kc)
        yA[kc] = *(const v2f*)(y0 + (size_t)(R + lo) * D_DIM + 4 * kc + 2 * hi);
    store_traj(out, yA, myScr, R, 0, lane, lo, hi);

    v2f kA[16], sumA[16], ytA[16];
#pragma unroll 1
    for (int step = 0; step < T_STEPS - 1; ++step) {
        float dt = ts[step + 1] - ts[step];
        float hdt = 0.5f * dt;

        eval_f(yA, kA, W1f, W2f, b1s, b2s, myScr, lane, lo, hi);   // k1
#pragma unroll
        for (int i = 0; i < 16; ++i) { sumA[i] = kA[i]; ytA[i] = yA[i] + kA[i] * hdt; }

        eval_f(ytA, kA, W1f, W2f, b1s, b2s, myScr, lane, lo, hi);  // k2
#pragma unroll
        for (int i = 0; i < 16; ++i) { sumA[i] += kA[i] * 2.0f; ytA[i] = yA[i] + kA[i] * hdt; }

        eval_f(ytA, kA, W1f, W2f, b1s, b2s, myScr, lane, lo, hi);  // k3
#pragma unroll
        for (int i = 0; i < 16; ++i) { sumA[i] += kA[i] * 2.0f; ytA[i] = yA[i] + kA[i] * dt; }

        eval_f(ytA, kA, W1f, W2f, b1s, b2s, myScr, lane, lo, hi);  // k4
        float s6 = dt * (1.0f / 6.0f);
#pragma unroll
        for (int i = 0; i < 16; ++i) yA[i] += (sumA[i] + kA[i]) * s6;

        store_traj(out, yA, myScr, R, step + 1, lane, lo, hi);
    }
}

extern "C" void kernel_launch(void* const* d_in, const int* in_sizes, int n_in,
                              void* d_out, int out_size, void* d_ws, size_t ws_size,
                              hipStream_t stream) {
    const float* y0 = (const float*)d_in[0];
    const float* ts = (const float*)d_in[1];
    const float* W1 = (const float*)d_in[2];
    const float* b1 = (const float*)d_in[3];
    const float* W2 = (const float*)d_in[4];
    const float* b2 = (const float*)d_in[5];
    float* out = (float*)d_out;

    // 2 waves/block * 16 rows/wave = 32 rows/block; 128 blocks cover N=4096.
    size_t shmem = (size_t)(D_DIM * H_DIM + H_DIM * D_DIM + H_DIM + D_DIM + 2 * 1024) * sizeof(float);
    node_rk4_kernel<<<dim3(N_PTS / 32), dim3(64), shmem, stream>>>(y0, ts, W1, b1, W2, b2, out);
}